// RoPEAttention_57191784514006
// MI455X (gfx1250) — compile-verified
//
#include <hip/hip_runtime.h>
#include <hip/hip_bf16.h>
#include <math.h>

typedef __attribute__((ext_vector_type(2))) float v2f;
typedef __attribute__((ext_vector_type(8))) float v8f;
typedef __attribute__((address_space(3))) float lds_f32;

#define B_DIM 2
#define S_DIM 2048
#define D_DIM 1024
#define H_DIM 16
#define HD_DIM 64
#define EPSF 1e-6f

// ---------------- CDNA5 async global->LDS helpers ----------------
// GLOBAL_LOAD_ASYNC_TO_LDS_B128: per-lane 16B copy, tracked with ASYNCcnt.
// VDST = VGPR holding LDS byte offset (addrspace(3) pointer), VADDR = 64-bit
// global address pair, GV mode (saddr = off).
__device__ __forceinline__ void async_cp16(float* dst_lds, const float* src) {
    lds_f32* p3 = (lds_f32*)dst_lds;           // addrspacecast -> 32-bit LDS offset
    asm volatile("global_load_async_to_lds_b128 %0, %1, off"
                 :: "v"(p3), "v"(src) : "memory");
}
#define WAIT_ASYNCCNT(N) asm volatile("s_wait_asynccnt %0" :: "n"(N) : "memory")

// ---------------- Generic batched WMMA-f32 GEMM ----------------
// C = alpha * (A @ B or A @ B^T) + bias
// Block tile BM=128 x BN=64, KB=32; 256 threads = 8 waves (wave32).
// Waves: 4 row-slabs x 2 col-slabs; each wave owns 32x32 of C
// (2x2 grid of 16x16 f32 WMMA accumulators -> 4 WMMA per K-step).
// Double-buffered LDS fed by async global->LDS copies (ASYNCcnt pipelined).
#define BM 128
#define BN 64
#define KB 32
#define LDAS 36   // 144B row stride: 16B aligned, conflict-free frag reads
#define LDBS 72   // 288B row stride: +2 rows => +16 banks, conflict-free

__global__ __launch_bounds__(256)
void gemm_f32_wmma(const float* __restrict__ A, const float* __restrict__ Bm,
                   const float* __restrict__ bias, float* __restrict__ C,
                   int M, int N, int K,
                   long lda, long ldb, long ldc,
                   int transB, float alpha, int Hsub,
                   long sAb, long sAh, long sBb, long sBh, long sCb, long sCh)
{
    __shared__ __align__(16) float As[2][BM * LDAS];
    __shared__ __align__(16) float Bs[2][KB * LDBS];

    const int tid      = threadIdx.x;
    const int lane     = tid & 31;
    const int w        = tid >> 5;
    const int laneMod  = lane & 15;
    const int laneHalf = lane >> 4;
    const int waveRow  = w & 3;   // 0..3 -> 32-row slabs
    const int waveCol  = w >> 2;  // 0..1 -> 32-col slabs

    const int n0 = blockIdx.x * BN;
    const int m0 = blockIdx.y * BM;
    const int z  = blockIdx.z;
    const int zb = z / Hsub;
    const int zh = z % Hsub;

    const float* Ab = A  + (size_t)zb * sAb + (size_t)zh * sAh;
    const float* Bb = Bm + (size_t)zb * sBb + (size_t)zh * sBh;
    float*       Cb = C  + (size_t)zb * sCb + (size_t)zh * sCh;

    // stage tile k0 into LDS buffer `buf`
    auto stage = [&](int k0, int buf) {
        // A tile: BM x KB = 4096 floats = 1024 x 16B, 4 per thread (async)
        #pragma unroll
        for (int i = 0; i < 4; ++i) {
            const int idx = tid + i * 256;
            const int r   = idx >> 3;            // 0..127
            const int c   = (idx & 7) << 2;      // 0,4,..,28
            async_cp16(&As[buf][r * LDAS + c],
                       Ab + (size_t)(m0 + r) * lda + k0 + c);
        }
        if (!transB) {
            // B tile: KB x BN = 2048 floats = 512 x 16B, 2 per thread (async)
            #pragma unroll
            for (int i = 0; i < 2; ++i) {
                const int idx = tid + i * 256;
                const int kk  = idx >> 4;        // 0..31
                const int c   = (idx & 15) << 2; // 0,4,..,60
                async_cp16(&Bs[buf][kk * LDBS + c],
                           Bb + (size_t)(k0 + kk) * ldb + n0 + c);
            }
        } else {
            // B stored N x K; transpose while staging: Bs[kk][n] = B[n0+n][k0+kk]
            #pragma unroll
            for (int i = 0; i < 2; ++i) {
                const int idx = tid + i * 256;
                const int nn  = idx >> 3;        // 0..63
                const int c   = (idx & 7) << 2;  // 0,4,..,28
                const float4 v = *(const float4*)(Bb + (size_t)(n0 + nn) * ldb + k0 + c);
                Bs[buf][(c + 0) * LDBS + nn] = v.x;
                Bs[buf][(c + 1) * LDBS + nn] = v.y;
                Bs[buf][(c + 2) * LDBS + nn] = v.z;
                Bs[buf][(c + 3) * LDBS + nn] = v.w;
            }
        }
    };

    v8f acc00 = {}, acc01 = {}, acc10 = {}, acc11 = {};

    stage(0, 0);
    int buf = 0;
    for (int k0 = 0; k0 < K; k0 += KB, buf ^= 1) {
        const bool hasNext = (k0 + KB) < K;
        if (hasNext) stage(k0 + KB, buf ^ 1);    // overlap next tile with compute

        // wait for *this* tile's async copies (next tile's stay in flight)
        if (hasNext) { if (transB) WAIT_ASYNCCNT(4); else WAIT_ASYNCCNT(6); }
        else         { WAIT_ASYNCCNT(0); }
        __syncthreads();                          // all waves' tile visible

        const float* Ap = &As[buf][0];
        const float* Bp = &Bs[buf][0];
        #pragma unroll
        for (int kk = 0; kk < KB; kk += 4) {
            v2f a0, a1, b0, b1;
            const int ar = waveRow * 32 + laneMod;
            a0.x = Ap[(ar)      * LDAS + kk + 2 * laneHalf];
            a0.y = Ap[(ar)      * LDAS + kk + 2 * laneHalf + 1];
            a1.x = Ap[(ar + 16) * LDAS + kk + 2 * laneHalf];
            a1.y = Ap[(ar + 16) * LDAS + kk + 2 * laneHalf + 1];
            const int bc = waveCol * 32 + laneMod;
            b0.x = Bp[(kk + 2 * laneHalf)     * LDBS + bc];
            b0.y = Bp[(kk + 2 * laneHalf + 1) * LDBS + bc];
            b1.x = Bp[(kk + 2 * laneHalf)     * LDBS + bc + 16];
            b1.y = Bp[(kk + 2 * laneHalf + 1) * LDBS + bc + 16];
            acc00 = __builtin_amdgcn_wmma_f32_16x16x4_f32(false, a0, false, b0, (short)0, acc00, false, false);
            acc01 = __builtin_amdgcn_wmma_f32_16x16x4_f32(false, a0, false, b1, (short)0, acc01, false, false);
            acc10 = __builtin_amdgcn_wmma_f32_16x16x4_f32(false, a1, false, b0, (short)0, acc10, false, false);
            acc11 = __builtin_amdgcn_wmma_f32_16x16x4_f32(false, a1, false, b1, (short)0, acc11, false, false);
        }
        __syncthreads();                          // buf free for tile k0+2*KB
    }

    // ---- epilogue: c[i] -> row = i + 8*laneHalf, col = laneMod ----
    const int rowBase = m0 + waveRow * 32 + 8 * laneHalf;
    const int col0 = n0 + waveCol * 32 + laneMod;
    const int col1 = col0 + 16;
    const float bias0 = bias ? bias[col0] : 0.0f;
    const float bias1 = bias ? bias[col1] : 0.0f;
    #pragma unroll
    for (int i = 0; i < 8; ++i) {
        const size_t r0 = (size_t)(rowBase + i) * ldc;
        const size_t r1 = (size_t)(rowBase + 16 + i) * ldc;
        Cb[r0 + col0] = alpha * acc00[i] + bias0;
        Cb[r0 + col1] = alpha * acc01[i] + bias1;
        Cb[r1 + col0] = alpha * acc10[i] + bias0;
        Cb[r1 + col1] = alpha * acc11[i] + bias1;
    }
}

// ---------------- RoPE + RMSNorm, in place on q/k sections ----------------
// Rows ordered (b, s, sec, h); sec 0 = q (wq), 1 = k (wk). hd = 64 lanes/row.
__global__ __launch_bounds__(256)
void rope_rmsnorm(float* __restrict__ qkv,
                  const float* __restrict__ wq, const float* __restrict__ wk)
{
    __shared__ float red[4][64];
    const int tid = threadIdx.x;
    const int rl  = tid >> 6;    // 0..3: row within block
    const int d   = tid & 63;    // element within head dim

    long r = (long)blockIdx.x * 4 + rl;
    const int h = (int)(r % H_DIM);
    long t = r / H_DIM;
    const int sec = (int)(t & 1);
    t >>= 1;
    const int s = (int)(t % S_DIM);
    const int b = (int)(t / S_DIM);

    float* p = qkv + ((size_t)b * S_DIM + s) * (3 * D_DIM)
                   + (size_t)sec * D_DIM + (size_t)h * HD_DIM;

    // RoPE: inv_freq[j] = 10000^(-j/32); angle repeats over the two halves
    const int j = d & 31;
    const float inv_freq = powf(10000.0f, -(float)j * (1.0f / 32.0f));
    const float ang = (float)s * inv_freq;
    const float cs = cosf(ang);
    const float sn = sinf(ang);
    const float x  = p[d];
    const float xr = (d < 32) ? -p[d + 32] : p[d - 32];
    const float y  = x * cs + xr * sn;

    // RMS norm with eps added to the norm (not the variance)
    red[rl][d] = y * y;
    __syncthreads();
    if (d < 32) {
        float v = red[rl][d] + red[rl][d + 32];
        for (int off = 16; off > 0; off >>= 1) v += __shfl_down(v, off, 32);
        if (d == 0) red[rl][0] = v;
    }
    __syncthreads();
    const float norm = sqrtf(red[rl][0] * (1.0f / 64.0f));
    const float wgt  = (sec == 0 ? wq : wk)[d];
    p[d] = y / (norm + EPSF) * wgt;
}

// ---------------- row softmax over attn_weights (rows of length S) ----------
__global__ __launch_bounds__(256)
void softmax_rows(float* __restrict__ attnw, const unsigned char* __restrict__ mask)
{
    __shared__ float red[256];
    const long r  = blockIdx.x;             // (b*H + h)*S + s
    const int tid = threadIdx.x;
    float* p = attnw + (size_t)r * S_DIM;
    const unsigned char* mrow = mask + (size_t)(r % S_DIM) * S_DIM;

    const int base = tid * 8;
    float vals[8];
    {
        const float4 v0 = *(const float4*)(p + base);
        const float4 v1 = *(const float4*)(p + base + 4);
        vals[0] = v0.x; vals[1] = v0.y; vals[2] = v0.z; vals[3] = v0.w;
        vals[4] = v1.x; vals[5] = v1.y; vals[6] = v1.z; vals[7] = v1.w;
    }
    float mx = -INFINITY;
    #pragma unroll
    for (int i = 0; i < 8; ++i) {
        if (!mrow[base + i]) vals[i] = -INFINITY;
        mx = fmaxf(mx, vals[i]);
    }
    red[tid] = mx; __syncthreads();
    for (int s = 128; s > 0; s >>= 1) {
        if (tid < s) red[tid] = fmaxf(red[tid], red[tid + s]);
        __syncthreads();
    }
    mx = red[0];
    __syncthreads();

    float sum = 0.0f;
    #pragma unroll
    for (int i = 0; i < 8; ++i) {
        const float e = (vals[i] == -INFINITY) ? 0.0f : __expf(vals[i] - mx);
        vals[i] = e;
        sum += e;
    }
    red[tid] = sum; __syncthreads();
    for (int s = 128; s > 0; s >>= 1) {
        if (tid < s) red[tid] += red[tid + s];
        __syncthreads();
    }
    sum = red[0];
    const float inv = (sum > 0.0f) ? (1.0f / sum) : 0.0f;  // nan_to_num path

    float4 o0, o1;
    o0.x = vals[0] * inv; o0.y = vals[1] * inv; o0.z = vals[2] * inv; o0.w = vals[3] * inv;
    o1.x = vals[4] * inv; o1.y = vals[5] * inv; o1.z = vals[6] * inv; o1.w = vals[7] * inv;
    *(float4*)(p + base)     = o0;
    *(float4*)(p + base + 4) = o1;
}

// ---------------- host-side orchestration ----------------
extern "C" void kernel_launch(void* const* d_in, const int* in_sizes, int n_in,
                              void* d_out, int out_size, void* d_ws, size_t ws_size,
                              hipStream_t stream) {
    (void)in_sizes; (void)n_in; (void)out_size; (void)ws_size;

    const float* x     = (const float*)d_in[0];
    const float* Wqkv  = (const float*)d_in[1];
    const float* bqkv  = (const float*)d_in[2];
    const float* Wproj = (const float*)d_in[3];
    const float* bproj = (const float*)d_in[4];
    const float* wq    = (const float*)d_in[5];
    const float* wk    = (const float*)d_in[6];
    const unsigned char* mask = (const unsigned char*)d_in[7];

    float* out   = (float*)d_out;                                   // (B,S,D)
    float* attnw = out + (size_t)B_DIM * S_DIM * D_DIM;             // (B,H,S,S)
    float* qkv   = (float*)d_ws;                                    // (B,S,3D)

    const int  BS   = B_DIM * S_DIM;          // 4096
    const long D3   = 3L * D_DIM;             // 3072
    const long sBH  = (long)S_DIM * D3;       // per-batch stride within qkv
    const long sSS  = (long)S_DIM * S_DIM;    // one attention map

    dim3 blk(256);

    // 1) qkv = x @ Wqkv + bqkv        (4096 x 3072 x 1024)
    gemm_f32_wmma<<<dim3(3 * D_DIM / BN, BS / BM, 1), blk, 0, stream>>>(
        x, Wqkv, bqkv, qkv, BS, 3 * D_DIM, D_DIM,
        D_DIM, 3L * D_DIM, 3L * D_DIM, /*transB=*/0, 1.0f, /*Hsub=*/1,
        0, 0, 0, 0, 0, 0);

    // 2) RoPE + RMSNorm in place on q and k sections
    const long rope_rows = (long)B_DIM * S_DIM * 2 * H_DIM;          // 131072
    rope_rmsnorm<<<dim3((unsigned)(rope_rows / 4)), blk, 0, stream>>>(qkv, wq, wk);

    // 3) scores = q @ k^T * hd^-0.5   per (b,h): 2048 x 2048 x 64
    gemm_f32_wmma<<<dim3(S_DIM / BN, S_DIM / BM, B_DIM * H_DIM), blk, 0, stream>>>(
        qkv /*q*/, qkv + D_DIM /*k*/, nullptr, attnw,
        S_DIM, S_DIM, HD_DIM,
        D3, D3, S_DIM, /*transB=*/1, 0.125f, /*Hsub=*/H_DIM,
        sBH, HD_DIM, sBH, HD_DIM, (long)H_DIM * sSS, sSS);

    // 4) softmax over each row of attn_weights (mask-aware, nan_to_num)
    softmax_rows<<<dim3((unsigned)((long)B_DIM * H_DIM * S_DIM)), blk, 0, stream>>>(
        attnw, mask);

    // 5) attn_out = attn_w @ v        per (b,h): 2048 x 64 x 2048
    //    written into the (dead) q slots of qkv -> layout (B,S,H,hd)=(B,S,D)
    gemm_f32_wmma<<<dim3(HD_DIM / BN, S_DIM / BM, B_DIM * H_DIM), blk, 0, stream>>>(
        attnw, qkv + 2L * D_DIM /*v*/, nullptr, qkv,
        S_DIM, HD_DIM, S_DIM,
        S_DIM, D3, D3, /*transB=*/0, 1.0f, /*Hsub=*/H_DIM,
        (long)H_DIM * sSS, sSS, sBH, HD_DIM, sBH, HD_DIM);

    // 6) output = attn_out @ Wproj + bproj   (4096 x 1024 x 1024)
    gemm_f32_wmma<<<dim3(D_DIM / BN, BS / BM, 1), blk, 0, stream>>>(
        qkv, Wproj, bproj, out, BS, D_DIM, D_DIM,
        D3, D_DIM, D_DIM, /*transB=*/0, 1.0f, /*Hsub=*/1,
        0, 0, 0, 0, 0, 0);
}